// Reservoir_72335839199403
// MI455X (gfx1250) — compile-verified
//
#include <hip/hip_runtime.h>
#include <math.h>

#define DT 0.01f
#define PBLK 256          // threads per persistent block (8 wave32)
#define ROWS_PER_BLK 32   // N / ROWS_PER_BLK = grid of the persistent kernel

typedef float v2f __attribute__((ext_vector_type(2)));
typedef float v8f __attribute__((ext_vector_type(8)));

// ---------------------------------------------------------------------------
// Kernel 0: per-call init (graph-replay safe): s0 <- state0, counter <- 0
// ---------------------------------------------------------------------------
__global__ void rsvr_init(const float* __restrict__ state0,
                          float* __restrict__ s0,
                          unsigned* __restrict__ counter, int N) {
    int i = blockIdx.x * blockDim.x + threadIdx.x;
    if (i < N) s0[i] = state0[i];
    if (i == 0) *counter = 0u;
}

// ---------------------------------------------------------------------------
// Kernel 1: W_eff = W_rec + W_in @ W_out^T  via V_WMMA_F32_16X16X4_F32
//   W_in  : (N, K) row-major,  W_out : (N, K) row-major
//   W_fb[m,n] = sum_k W_in[m,k] * W_out[n,k]
// One 16x16 output tile per wave; K looped in steps of 4, C chained.
// fp32 A-layout (ISA 7.12.2): lanes 0-15 hold M=0..15 with K={k0,k0+1} in
// VGPR0/1; lanes 16-31 hold K={k0+2,k0+3}. B mirrored (N across lanes).
// D: VGPR v -> row (v + 8*(lane>=16)), col lane%16.
// ---------------------------------------------------------------------------
__global__ void rsvr_weff_wmma(const float* __restrict__ W_in,
                               const float* __restrict__ W_out,
                               const float* __restrict__ W_rec,
                               float* __restrict__ W_eff,
                               int N, int K) {
    const int lane = threadIdx.x & 31;
    const int wave = threadIdx.x >> 5;
    const int half = lane >> 4;     // 0: lanes 0-15, 1: lanes 16-31
    const int lr   = lane & 15;
    const int n0 = blockIdx.x * 16;
    const int m0 = (blockIdx.y * 8 + wave) * 16;
    const int m  = m0 + lr;         // A row this lane carries
    const int n  = n0 + lr;         // B col this lane carries

    v8f c = {0.f, 0.f, 0.f, 0.f, 0.f, 0.f, 0.f, 0.f};
    for (int k0 = 0; k0 < K; k0 += 4) {
        const int ka = k0 + half * 2;
        v2f a, b;
        a.x = W_in [(size_t)m * K + ka];
        a.y = W_in [(size_t)m * K + ka + 1];
        b.x = W_out[(size_t)n * K + ka];
        b.y = W_out[(size_t)n * K + ka + 1];
        // (neg_a, A, neg_b, B, c_mod, C, reuse_a, reuse_b)
        c = __builtin_amdgcn_wmma_f32_16x16x4_f32(false, a, false, b,
                                                  (short)0, c, false, false);
    }
#pragma unroll
    for (int v = 0; v < 8; ++v) {
        const int row = m0 + half * 8 + v;
        const int col = n0 + lr;
        const size_t idx = (size_t)row * N + col;
        W_eff[idx] = W_rec[idx] + c[v];
    }
}

// ---------------------------------------------------------------------------
// Kernel 2: persistent reservoir recurrence with device-wide barrier.
//   Phase 1: ntr steps with W_rec (free-running, no record)
//   Phase 2: nt  steps with W_eff (record pre-step state into out)
// Each block owns ROWS_PER_BLK rows; 4 rows per wave; wave32 shfl reduction.
// Ping-pong state buffers avoid a second barrier per step.
// ---------------------------------------------------------------------------
__global__ void rsvr_run(const float* __restrict__ W_rec,
                         const float* __restrict__ W_eff,
                         float* __restrict__ s_a,
                         float* __restrict__ s_b,
                         float* __restrict__ out,          // (nt, N)
                         unsigned* __restrict__ counter,
                         const int* __restrict__ nt_transient_p,
                         int N, int nt) {
    __shared__ __align__(16) float s_lds[2048];            // N <= 2048
    const int tid  = threadIdx.x;
    const int lane = tid & 31;
    const int wave = tid >> 5;
    const int nblk = gridDim.x;
    const int row0 = blockIdx.x * ROWS_PER_BLK;
    const int ntr  = *nt_transient_p;

    float* cur = s_a;
    float* nxt = s_b;
    unsigned target = 0;

    for (int step = 0; step < ntr + nt; ++step) {
        // Stage full current state into LDS (float4, fully coalesced).
        for (int i = tid; i < (N >> 2); i += PBLK)
            ((float4*)s_lds)[i] = ((const float4*)cur)[i];
        __syncthreads();

        const bool rec = (step >= ntr);
        if (rec && tid < ROWS_PER_BLK)   // each block records only its slice
            out[(size_t)(step - ntr) * N + row0 + tid] = s_lds[row0 + tid];

        const float* __restrict__ W = rec ? W_eff : W_rec;

#pragma unroll
        for (int rr = 0; rr < 4; ++rr) {
            const int row = row0 + wave * 4 + rr;
            const float* __restrict__ wr = W + (size_t)row * N;
            float acc = 0.f;
            // lane covers k = lane*4 + 128*i : 32 lanes * float4 = 512B/sweep
            for (int k = lane * 4; k < N; k += 128) {
                __builtin_prefetch(wr + N + k, 0, 0);      // next row, L2->L0
                const float4 wv = *(const float4*)(wr + k);
                const float4 sv = *(const float4*)(s_lds + k);
                acc = fmaf(wv.x, sv.x, acc);
                acc = fmaf(wv.y, sv.y, acc);
                acc = fmaf(wv.z, sv.z, acc);
                acc = fmaf(wv.w, sv.w, acc);
            }
            // wave32 butterfly reduction
            for (int m = 16; m > 0; m >>= 1)
                acc += __shfl_xor(acc, m, 32);
            if (lane == 0) {
                const float s = s_lds[row];
                nxt[row] = s + DT * (-s + tanhf(acc));     // Euler step
            }
        }

        // ---- device-wide barrier: monotonic counter, release/acquire ----
        __syncthreads();                 // all waves finished writing nxt
        target += (unsigned)nblk;
        if (tid == 0) {
            __hip_atomic_fetch_add(counter, 1u, __ATOMIC_RELEASE,
                                   __HIP_MEMORY_SCOPE_AGENT);
            while (__hip_atomic_load(counter, __ATOMIC_ACQUIRE,
                                     __HIP_MEMORY_SCOPE_AGENT) < target) {
                __builtin_amdgcn_s_sleep(1);
            }
        }
        __syncthreads();
        __threadfence();                 // every wave invalidates its L0
        float* t = cur; cur = nxt; nxt = t;
    }
}

// ---------------------------------------------------------------------------
// inputs (setup_inputs order): x(nt,n_in), W_in(N,n_in), W_out(N,n_in),
//                              W_rec(N,N), state0(N), nt_transient(1)
// output: s_rec (nt, N) float32.  x is unused by the reference math.
// ---------------------------------------------------------------------------
extern "C" void kernel_launch(void* const* d_in, const int* in_sizes, int n_in_args,
                              void* d_out, int out_size, void* d_ws, size_t ws_size,
                              hipStream_t stream) {
    const float* W_in   = (const float*)d_in[1];
    const float* W_out  = (const float*)d_in[2];
    const float* W_rec  = (const float*)d_in[3];
    const float* state0 = (const float*)d_in[4];
    const int*   ntr_p  = (const int*)d_in[5];

    const int N    = in_sizes[4];              // 2048
    const int K    = in_sizes[1] / N;          // 128
    const int nt   = in_sizes[0] / K;          // 8192

    // workspace layout
    float*    W_eff   = (float*)d_ws;                                  // N*N f32
    float*    s0      = (float*)((char*)d_ws + (size_t)N * N * sizeof(float));
    float*    s1      = s0 + N;
    unsigned* counter = (unsigned*)(s1 + N);

    // 0) init state + barrier counter (part of the graph: replay-safe)
    rsvr_init<<<(N + 255) / 256, 256, 0, stream>>>(state0, s0, counter, N);

    // 1) W_eff = W_rec + W_in @ W_out^T  (WMMA fp32 GEMM, one-time)
    dim3 gemm_grid(N / 16, N / 128);           // 16-wide tiles x (8 waves * 16 rows)
    rsvr_weff_wmma<<<gemm_grid, 256, 0, stream>>>(W_in, W_out, W_rec, W_eff, N, K);

    // 2) persistent recurrence (64 blocks: small grid => cheap device barrier)
    const int nblk = N / ROWS_PER_BLK;         // 64
    rsvr_run<<<nblk, PBLK, 0, stream>>>(W_rec, W_eff, s0, s1,
                                        (float*)d_out, counter, ntr_p, N, nt);
}